// FasterRCNN_46634754900571
// MI455X (gfx1250) — compile-verified
//
#include <hip/hip_runtime.h>
#include <hip/hip_bf16.h>
#include <math.h>

#define NDET 6000
#define NCLS 81
#define NPAD 8192            // next pow2 >= NDET for bitonic sort
#define SUPW ((NDET + 31) / 32)
#define NMS_THR 0.3f

typedef __attribute__((ext_vector_type(4))) unsigned int u32x4;
typedef __attribute__((ext_vector_type(8))) int          i32x8;
typedef __attribute__((ext_vector_type(4))) int          i32x4;

#define HAVE_TDM (__has_builtin(__builtin_amdgcn_tensor_load_to_lds) && \
                  __has_builtin(__builtin_amdgcn_tensor_store_from_lds) && \
                  __has_builtin(__builtin_amdgcn_s_wait_tensorcnt))

#if HAVE_TDM
// Build a D# for a contiguous 1-row 2-D tile of n8 8-byte units and issue
// a TDM transfer. Group0: count=1 | lds_addr | global_addr[56:0] | type=2.
// Group1: data_size=3 (8B); tensor_dim0 = tile_dim0 = dim0_stride = n8;
//         tensor_dim1 = tile_dim1 = 1. Groups 2/3 unused (2-D tensor).
__device__ __forceinline__ void tdm_load_to_lds(unsigned int lds,
                                                unsigned long long ga, int n8)
{
    u32x4 g0 = { 1u, lds, (unsigned)ga,
                 (((unsigned)(ga >> 32)) & 0x01FFFFFFu) | 0x80000000u };
    i32x8 g1 = { (int)0x00030000,
                 (int)((unsigned)n8 << 16),   // tensor_dim0[15:0] @ bits[63:48]
                 (int)(1u << 16),             // tensor_dim1[15:0] @ bits[95:80]
                 (int)((unsigned)n8 << 16),   // tile_dim0 @ bits[127:112]
                 1,                           // tile_dim1 @ bits[143:128]
                 n8,                          // tensor_dim0_stride low 32
                 0, 0 };
    i32x4 z4 = { 0, 0, 0, 0 };
    i32x8 z8 = { 0, 0, 0, 0, 0, 0, 0, 0 };
    __builtin_amdgcn_tensor_load_to_lds(g0, g1, z4, z4, z8, 0);
}

__device__ __forceinline__ void tdm_store_from_lds(unsigned int lds,
                                                   unsigned long long ga, int n8)
{
    u32x4 g0 = { 1u, lds, (unsigned)ga,
                 (((unsigned)(ga >> 32)) & 0x01FFFFFFu) | 0x80000000u };
    i32x8 g1 = { (int)0x00030000,
                 (int)((unsigned)n8 << 16),
                 (int)(1u << 16),
                 (int)((unsigned)n8 << 16),
                 1,
                 n8,
                 0, 0 };
    i32x4 z4 = { 0, 0, 0, 0 };
    i32x8 z8 = { 0, 0, 0, 0, 0, 0, 0, 0 };
    __builtin_amdgcn_tensor_store_from_lds(g0, g1, z4, z4, z8, 0);
}
#endif

// ---------------- workspace layout (bytes) ----------------
#define WS_KEYS    0                           // 8192 * 8        = 65536
#define WS_BOXES   65536                       // 6000 * 16       = 96000
#define WS_SCORE   (WS_BOXES + 96000)          // 6000 * 4
#define WS_CLS     (WS_SCORE + 24000)          // 6000 * 4
#define WS_VALID   (WS_CLS + 24000)            // 6000 * 1
#define WS_SBOXES  (WS_VALID + 6000 + 16)      // 16B aligned; 6000 * 16
#define WS_SVALID  (WS_SBOXES + 96000)         // 6000
#define WS_KEEP    (WS_SVALID + 6000)          // 6000

// =====================================================================
// 1) decode: one wave32 per proposal row. Softmax/argmax over 81 classes
//    via butterfly shuffles, per-class regression decode, clip, sort key.
// =====================================================================
__global__ __launch_bounds__(256) void decode_kernel(
    const float* __restrict__ proposals, const float* __restrict__ reg,
    const float* __restrict__ clss, const float* __restrict__ stds,
    const int* __restrict__ img_size,
    unsigned long long* __restrict__ keys,
    float4* __restrict__ boxes, float* __restrict__ score,
    int* __restrict__ cls_out, unsigned char* __restrict__ valid)
{
    const int lane = threadIdx.x & 31;
    const int row  = blockIdx.x * (blockDim.x >> 5) + (threadIdx.x >> 5);
    if (row >= NPAD) return;
    if (row >= NDET) {                 // pad entries sort strictly last
        if (lane == 0) keys[row] = ~0ull;
        return;
    }

    const float* cl = clss + row * NCLS;
    float v0 = cl[lane];
    float v1 = cl[lane + 32];
    float v2 = (lane + 64 < NCLS) ? cl[lane + 64] : -INFINITY;

    float bestV = v0; int bestI = lane;
    if (v1 > bestV) { bestV = v1; bestI = lane + 32; }
    if (v2 > bestV) { bestV = v2; bestI = lane + 64; }

    // butterfly argmax, tie-break = smallest index (matches jnp.argmax)
    #pragma unroll
    for (int off = 16; off > 0; off >>= 1) {
        float oV = __shfl_xor(bestV, off, 32);
        int   oI = __shfl_xor(bestI, off, 32);
        if (oV > bestV || (oV == bestV && oI < bestI)) { bestV = oV; bestI = oI; }
    }

    // softmax denominator (stable); score = exp(max-max)/sum = 1/sum
    float se = expf(v0 - bestV) + expf(v1 - bestV);
    if (lane + 64 < NCLS) se += expf(v2 - bestV);
    #pragma unroll
    for (int off = 16; off > 0; off >>= 1) se += __shfl_xor(se, off, 32);

    if (lane == 0) {
        const int  c   = bestI;
        const bool vld = (c != 0);
        float4 p = ((const float4*)proposals)[row];
        float w  = p.z - p.x, h = p.w - p.y;
        float cx = p.x + 0.5f * w, cy = p.y + 0.5f * h;

        const float* r = reg + (size_t)row * (NCLS * 4) + c * 4;
        float t0 = r[0] * stds[0], t1 = r[1] * stds[1];
        float t2 = r[2] * stds[2], t3 = r[3] * stds[3];

        float px = cx + w * t0, py = cy + h * t1;
        float pw = w * expf(t2), ph = h * expf(t3);
        float hi = (float)(img_size[0] - 1);

        float x1 = fminf(fmaxf(px - 0.5f * pw, 0.0f), hi);
        float y1 = fminf(fmaxf(py - 0.5f * ph, 0.0f), hi);
        float x2 = fminf(fmaxf(px + 0.5f * pw, 0.0f), hi);
        float y2 = fminf(fmaxf(py + 0.5f * ph, 0.0f), hi);

        boxes[row]   = make_float4(x1, y1, x2, y2);
        float sc     = 1.0f / se;
        score[row]   = sc;
        cls_out[row] = c;
        valid[row]   = vld ? 1 : 0;

        // composite key: score descending, stable index tie-break
        float ks = vld ? sc : -INFINITY;
        unsigned u = __float_as_uint(ks);
        u = (u & 0x80000000u) ? ~u : (u | 0x80000000u);   // monotone ascending
        unsigned kd = ~u;                                  // descending
        keys[row] = ((unsigned long long)kd << 32) | (unsigned)row;
    }
}

// =====================================================================
// 2) bitonic sort of 8192 u64 keys in 64 KB LDS (one workgroup).
//    Keys are DMA'd in and out with the Tensor Data Mover.
// =====================================================================
__global__ __launch_bounds__(1024) void sort_kernel(unsigned long long* __restrict__ keys)
{
    __shared__ unsigned long long s[NPAD];   // 64 KB
    const int tid = threadIdx.x;

#if HAVE_TDM
    if (tid < 32) {   // wave 0: one TDM DMA, 8192 x 8B -> LDS
        tdm_load_to_lds((unsigned int)(uintptr_t)&s[0],
                        (unsigned long long)(uintptr_t)keys, NPAD);
        __builtin_amdgcn_s_wait_tensorcnt(0);
    }
#else
    for (int i = tid; i < NPAD; i += 1024) s[i] = keys[i];
#endif
    __syncthreads();

    for (int k = 2; k <= NPAD; k <<= 1) {
        for (int j = k >> 1; j > 0; j >>= 1) {
            for (int t = tid; t < NPAD / 2; t += 1024) {
                int i   = 2 * t - (t & (j - 1));
                int ixj = i + j;
                bool asc = ((i & k) == 0);
                unsigned long long a = s[i], b = s[ixj];
                if ((a > b) == asc) { s[i] = b; s[ixj] = a; }
            }
            __syncthreads();
        }
    }

#if HAVE_TDM
    if (tid < 32) {   // wave 0: one TDM DMA, LDS -> 8192 x 8B
        tdm_store_from_lds((unsigned int)(uintptr_t)&s[0],
                           (unsigned long long)(uintptr_t)keys, NPAD);
        __builtin_amdgcn_s_wait_tensorcnt(0);
    }
#else
    for (int i = tid; i < NPAD; i += 1024) keys[i] = s[i];
#endif
}

// =====================================================================
// 3) gather boxes/valid into sorted order
// =====================================================================
__global__ void gather_kernel(const unsigned long long* __restrict__ keys,
                              const float4* __restrict__ boxes,
                              const unsigned char* __restrict__ valid,
                              float4* __restrict__ sboxes,
                              unsigned char* __restrict__ svalid)
{
    int i = blockIdx.x * blockDim.x + threadIdx.x;
    if (i >= NDET) return;
    int orig  = (int)(keys[i] & 0xFFFFFFFFull);
    sboxes[i] = boxes[orig];
    svalid[i] = valid[orig];
}

// =====================================================================
// 4) greedy NMS, single block. Sorted boxes (96 KB) staged into LDS via
//    the Tensor Data Mover; suppression bitset lives in LDS.
// =====================================================================
__global__ __launch_bounds__(1024) void nms_kernel(const float4* __restrict__ sboxes,
                                                   const unsigned char* __restrict__ svalid,
                                                   unsigned char* __restrict__ keep)
{
    __shared__ float4 lb[NDET];          // 96000 B — CDNA5 320KB LDS
    __shared__ unsigned int sup[SUPW];   // suppression bitset
    const int tid = threadIdx.x;

#if HAVE_TDM
    if (tid < 32) {   // wave 0: one TDM DMA, 12000 x 8B -> LDS
        tdm_load_to_lds((unsigned int)(uintptr_t)&lb[0],
                        (unsigned long long)(uintptr_t)sboxes, 2 * NDET);
        __builtin_amdgcn_s_wait_tensorcnt(0);
    }
#else
    for (int i = tid; i < NDET; i += 1024) lb[i] = sboxes[i];
#endif
    for (int i = tid; i < SUPW; i += 1024) sup[i] = 0;
    __syncthreads();

    for (int i = 0; i < NDET; ++i) {
        // sup[i] only written by iterations < i (they suppress j > i)
        bool suppressed = (sup[i >> 5] >> (i & 31)) & 1u;
        bool k = (svalid[i] != 0) && !suppressed;   // uniform across block
        if (tid == 0) keep[i] = k ? 1 : 0;
        if (k) {
            float4 a = lb[i];
            float areaA = (a.z - a.x) * (a.w - a.y);
            for (int j = i + 1 + tid; j < NDET; j += 1024) {
                float4 b = lb[j];
                float xx1 = fmaxf(a.x, b.x), yy1 = fmaxf(a.y, b.y);
                float xx2 = fminf(a.z, b.z), yy2 = fminf(a.w, b.w);
                float iw = fmaxf(xx2 - xx1, 0.0f), ih = fmaxf(yy2 - yy1, 0.0f);
                float inter = iw * ih;
                float areaB = (b.z - b.x) * (b.w - b.y);
                float iou = inter / (areaA + areaB - inter + 1e-9f);
                if (iou > NMS_THR) atomicOr(&sup[j >> 5], 1u << (j & 31));
            }
        }
        __syncthreads();
    }
}

// =====================================================================
// 5) scatter keep mask back to original order; write outputs
//    d_out = [boxes 6000*4 | score 6000 | cls 6000] as float
// =====================================================================
__global__ void scatter_kernel(const unsigned long long* __restrict__ keys,
                               const unsigned char* __restrict__ keep,
                               const float4* __restrict__ boxes,
                               const float* __restrict__ score,
                               const int* __restrict__ cls,
                               float* __restrict__ out)
{
    int i = blockIdx.x * blockDim.x + threadIdx.x;
    if (i >= NDET) return;
    int orig = (int)(keys[i] & 0xFFFFFFFFull);
    float m  = keep[i] ? 1.0f : 0.0f;
    float4 b = boxes[orig];
    out[orig * 4 + 0] = b.x * m;
    out[orig * 4 + 1] = b.y * m;
    out[orig * 4 + 2] = b.z * m;
    out[orig * 4 + 3] = b.w * m;
    out[NDET * 4 + orig] = score[orig] * m;
    out[NDET * 5 + orig] = (float)cls[orig] * m;
}

// =====================================================================
extern "C" void kernel_launch(void* const* d_in, const int* in_sizes, int n_in,
                              void* d_out, int out_size, void* d_ws, size_t ws_size,
                              hipStream_t stream)
{
    const float* proposals = (const float*)d_in[0];
    const float* reg       = (const float*)d_in[1];
    const float* clss      = (const float*)d_in[2];
    const float* stds      = (const float*)d_in[3];
    const int*   img_size  = (const int*)d_in[4];

    char* ws = (char*)d_ws;
    unsigned long long* keys = (unsigned long long*)(ws + WS_KEYS);
    float4*        boxes  = (float4*)(ws + WS_BOXES);
    float*         score  = (float*)(ws + WS_SCORE);
    int*           cls    = (int*)(ws + WS_CLS);
    unsigned char* valid  = (unsigned char*)(ws + WS_VALID);
    float4*        sboxes = (float4*)(ws + WS_SBOXES);
    unsigned char* svalid = (unsigned char*)(ws + WS_SVALID);
    unsigned char* keep   = (unsigned char*)(ws + WS_KEEP);

    float* out = (float*)d_out;

    // 1) decode + key build (8 waves/block, 8192 rows incl. pad)
    decode_kernel<<<NPAD / 8, 256, 0, stream>>>(proposals, reg, clss, stds, img_size,
                                                keys, boxes, score, cls, valid);
    // 2) sort keys (TDM in/out of LDS)
    sort_kernel<<<1, 1024, 0, stream>>>(keys);
    // 3) gather into sorted order
    gather_kernel<<<(NDET + 255) / 256, 256, 0, stream>>>(keys, boxes, valid, sboxes, svalid);
    // 4) greedy NMS (single block, TDM-staged LDS)
    nms_kernel<<<1, 1024, 0, stream>>>(sboxes, svalid, keep);
    // 5) scatter outputs
    scatter_kernel<<<(NDET + 255) / 256, 256, 0, stream>>>(keys, keep, boxes, score, cls, out);
}